// DirectNoDecompHyperbolicForecaster_14413910245624
// MI455X (gfx1250) — compile-verified
//
#include <hip/hip_runtime.h>
#include <math.h>

#define EPSF 1e-15f
#define MAXNF (1.0f - 1e-5f)
#define ATCF (1.0f - 1e-7f)

// shapes
#define BB_ 128
#define LL_ 336
#define FF_ 128
#define SEG_ 24
#define DD_ 128
#define HH_ 256
#define TT_ 8
#define NN_ 14

typedef float v2f __attribute__((ext_vector_type(2)));
typedef float v8f __attribute__((ext_vector_type(8)));

__device__ __forceinline__ v8f wmma4(v2f a, v2f b, v8f c) {
  // D = A(16x4 f32) x B(4x16 f32) + C(16x16 f32)  -> v_wmma_f32_16x16x4_f32
  return __builtin_amdgcn_wmma_f32_16x16x4_f32(false, a, false, b, (short)0, c,
                                               false, false);
}

__device__ __forceinline__ float wred(float v) {
#pragma unroll
  for (int o = 16; o > 0; o >>= 1) v += __shfl_xor(v, o, 32);
  return v;
}

// ---------------- Kernel 0: RevIN statistics -------------------------------
__global__ __launch_bounds__(128) void k_stats(const float* __restrict__ x,
                                               float* __restrict__ mu,
                                               float* __restrict__ stdv) {
  int b = blockIdx.x, f = threadIdx.x;
  const float* xp = x + (size_t)b * LL_ * FF_ + f;
  float s = 0.f, s2 = 0.f;
  for (int l = 0; l < LL_; ++l) {
    float v = xp[(size_t)l * FF_];
    s += v;
    s2 += v * v;
  }
  float m = s / (float)LL_;
  float var = fmaxf(s2 / (float)LL_ - m * m, 0.f);
  mu[b * FF_ + f] = m;
  stdv[b * FF_ + f] = sqrtf(var + 1e-5f);
}

// ---------------- Kernel 1: encode + vels + v_init (block = one bf) --------
__global__ __launch_bounds__(128) void k_encode(
    const float* __restrict__ x, const float* __restrict__ revin_w,
    const float* __restrict__ revin_b, const float* __restrict__ W_enc,
    const float* __restrict__ b_enc, const float* __restrict__ mu,
    const float* __restrict__ stdv, float* __restrict__ vinit,
    float* __restrict__ z0out) {
  __shared__ float sW[SEG_ * 128];  // W_enc [k][n]
  __shared__ float sA[16 * 24];     // xb tile [m][k], rows 14..15 zero
  __shared__ float sZ[16 * 128];    // z tile [m][n]
  __shared__ float rowsq[16];
  __shared__ float cx[13], cy[13];

  int t = threadIdx.x;
  int bf = blockIdx.x;
  int b = bf >> 7, f = bf & 127;
  int lane = t & 31, w = t >> 5;

  // stage W_enc (contiguous)
#pragma unroll
  for (int q = 0; q < SEG_; ++q) sW[q * 128 + t] = W_enc[q * 128 + t];

  // build A tile with RevIN normalization  (xb[bf, n, s] = xn[b, f, n*24+s])
  float m_ = mu[bf], sd = stdv[bf], rw = revin_w[f], rb = revin_b[f];
  for (int idx = t; idx < 384; idx += 128) {
    float v = 0.f;
    if (idx < LL_)
      v = (x[(size_t)b * LL_ * FF_ + (size_t)idx * FF_ + f] - m_) / sd * rw + rb;
    sA[idx] = v;
  }
  __syncthreads();

  // GEMM (M=16, K=24, N=128): 4 waves x 2 N-tiles, 6 K-steps of f32 WMMA
  int m16 = lane & 15, koff = (lane < 16) ? 0 : 2;
#pragma unroll
  for (int ti = 0; ti < 2; ++ti) {
    int n0 = (w + 4 * ti) * 16;
    v8f c = {};
#pragma unroll
    for (int kk = 0; kk < 6; ++kk) {
      int k = kk * 4 + koff;
      v2f a, bb;
      a.x = sA[m16 * 24 + k];
      a.y = sA[m16 * 24 + k + 1];
      bb.x = sW[k * 128 + n0 + m16];
      bb.y = sW[(k + 1) * 128 + n0 + m16];
      c = wmma4(a, bb, c);
    }
#pragma unroll
    for (int r = 0; r < 8; ++r) {
      int mm = r + ((lane < 16) ? 0 : 8);
      int nn = n0 + m16;
      sZ[mm * 128 + nn] = c[r] + b_enc[nn];
    }
  }
  __syncthreads();

  // z = projx(expmap0(u)) per row; scale = min(tanh(n), MAXN)/n
  for (int r = w; r < 16; r += 4) {
    float ss = 0.f;
#pragma unroll
    for (int j = 0; j < 4; ++j) {
      float v = sZ[r * 128 + lane * 4 + j];
      ss += v * v;
    }
    ss = wred(ss);
    float n = sqrtf(fmaxf(ss, EPSF));
    float sc = fminf(tanhf(n), MAXNF) / n;
#pragma unroll
    for (int j = 0; j < 4; ++j) sZ[r * 128 + lane * 4 + j] *= sc;
    if (lane == 0) rowsq[r] = ss * sc * sc;
  }
  __syncthreads();

  // vel_i = cx[i]*z_i + cy[i]*z_{i+1}  (logmap(z_i, z_{i+1}) in scalar form)
  for (int i = w; i < 13; i += 4) {
    float s = 0.f;
#pragma unroll
    for (int j = 0; j < 4; ++j)
      s += sZ[i * 128 + lane * 4 + j] * sZ[(i + 1) * 128 + lane * 4 + j];
    s = wred(s);
    if (lane == 0) {
      float x2 = rowsq[i], y2 = rowsq[i + 1];
      float A = 1.f - 2.f * s + y2;   // coeff on (-x)
      float Bq = 1.f - x2;            // coeff on y
      float den = fmaxf(1.f - 2.f * s + x2 * y2, EPSF);
      float num2 = A * A * x2 + Bq * Bq * y2 - 2.f * A * Bq * s;
      float un2 = fmaxf(num2 / (den * den), EPSF);
      float n = sqrtf(un2);
      float fac = fmaxf(1.f - x2, EPSF) * atanhf(fminf(n, ATCF)) / n;
      cx[i] = -fac * A / den;
      cy[i] = fac * Bq / den;
    }
  }
  __syncthreads();

  // v_init = mean_i( w_i * vel_i )  => linear combo of 14 z-rows
  float wv[13];
  {
    float p = 1.f;
#pragma unroll
    for (int i = 12; i >= 0; --i) {
      wv[i] = p;
      p *= 0.9f;
    }
  }
  float wsum = 0.f;
#pragma unroll
  for (int i = 0; i < 13; ++i) wsum += wv[i];
  float inv = 1.f / (wsum * 13.f);  // jnp.mean over 13 entries

  int d = t;  // 128 threads == D
  float acc = 0.f;
#pragma unroll
  for (int n = 0; n < 14; ++n) {
    float g = 0.f;
    if (n <= 12) g += wv[n] * inv * cx[n];
    if (n >= 1) g += wv[n - 1] * inv * cy[n - 1];
    acc += g * sZ[n * 128 + d];
  }
  vinit[(size_t)bf * 128 + d] = acc;
  z0out[(size_t)bf * 128 + d] = sZ[13 * 128 + d];
}

// ---------------- Kernel 2: v0t = projx(mobius_add(mobius_matvec(Wv,vi), eb))
__global__ __launch_bounds__(256) void k_v0t(const float* __restrict__ vinit,
                                             const float* __restrict__ W_vel,
                                             const float* __restrict__ b_vel,
                                             float* __restrict__ v0t) {
  __shared__ float sA[16 * 129];  // v_init rows (padded stride)
  __shared__ float sB[32 * 129];  // W_vel^T K-chunk; reused as mx tile
  __shared__ float ebuf[128];
  __shared__ float xn2[16];
  __shared__ float seb2;

  int t = threadIdx.x, lane = t & 31, w = t >> 5;
  int row0 = blockIdx.x * 16;

#pragma unroll
  for (int q = 0; q < 8; ++q) {
    int e = q * 256 + t;
    int m = e >> 7, k = e & 127;
    sA[m * 129 + k] = vinit[(size_t)(row0 + m) * 128 + k];
  }
  __syncthreads();

  for (int r = w; r < 16; r += 8) {  // |v_init row|^2
    float ss = 0.f;
#pragma unroll
    for (int j = 0; j < 4; ++j) {
      float v = sA[r * 129 + lane * 4 + j];
      ss += v * v;
    }
    ss = wred(ss);
    if (lane == 0) xn2[r] = ss;
  }
  if (w == 0) {  // eb = expmap0(b_vel)
    float bs = 0.f, bv[4];
#pragma unroll
    for (int j = 0; j < 4; ++j) {
      bv[j] = b_vel[lane * 4 + j];
      bs += bv[j] * bv[j];
    }
    bs = wred(bs);
    float n = sqrtf(fmaxf(bs, EPSF));
    float sc = tanhf(n) / n;
#pragma unroll
    for (int j = 0; j < 4; ++j) ebuf[lane * 4 + j] = bv[j] * sc;
    if (lane == 0) seb2 = bs * sc * sc;
  }

  // GEMM mx = v_init @ W_vel^T : B[k][n] = W_vel[n][k]
  int m16 = lane & 15, koff = (lane < 16) ? 0 : 2;
  int n0 = w * 16;
  v8f c = {};
  for (int kc = 0; kc < 4; ++kc) {
    __syncthreads();
#pragma unroll
    for (int q = 0; q < 16; ++q) {
      int e = q * 256 + t;
      int kk = e & 31, nn = e >> 5;
      sB[kk * 129 + nn] = W_vel[nn * 128 + kc * 32 + kk];
    }
    __syncthreads();
#pragma unroll
    for (int k2 = 0; k2 < 8; ++k2) {
      int kl = k2 * 4 + koff;
      int kg = kc * 32 + kl;
      v2f a, bb;
      a.x = sA[m16 * 129 + kg];
      a.y = sA[m16 * 129 + kg + 1];
      bb.x = sB[kl * 129 + n0 + m16];
      bb.y = sB[(kl + 1) * 129 + n0 + m16];
      c = wmma4(a, bb, c);
    }
  }
  __syncthreads();
  float* mx = sB;  // reuse chunk buffer as mx tile
#pragma unroll
  for (int r = 0; r < 8; ++r) {
    int mm = r + ((lane < 16) ? 0 : 8);
    mx[mm * 129 + n0 + m16] = c[r];
  }
  __syncthreads();

  for (int r = w; r < 16; r += 8) {
    float smx2 = 0.f, sme = 0.f, mvv[4], evv[4];
#pragma unroll
    for (int j = 0; j < 4; ++j) {
      mvv[j] = mx[r * 129 + lane * 4 + j];
      evv[j] = ebuf[lane * 4 + j];
      smx2 += mvv[j] * mvv[j];
      sme += mvv[j] * evv[j];
    }
    smx2 = wred(smx2);
    sme = wred(sme);
    float xn = sqrtf(fmaxf(xn2[r], EPSF));
    float mxn = sqrtf(fmaxf(smx2, EPSF));
    float ms = tanhf(mxn / xn * atanhf(fminf(xn, ATCF))) / mxn;  // mw = ms*mx
    float mw2 = ms * ms * smx2;
    float s2 = ms * sme;  // dot(mw, eb)
    float eb2v = seb2;
    float A = 1.f + 2.f * s2 + eb2v;
    float Bq = 1.f - mw2;
    float den = fmaxf(1.f + 2.f * s2 + mw2 * eb2v, EPSF);
    float res2 = (A * A * mw2 + 2.f * A * Bq * s2 + Bq * Bq * eb2v) / (den * den);
    float rn = sqrtf(fmaxf(res2, EPSF));
    float ps = (rn > MAXNF) ? (MAXNF / rn) : 1.f;
    float cM = A / den * ps * ms;
    float cE = Bq / den * ps;
#pragma unroll
    for (int j = 0; j < 4; ++j)
      v0t[(size_t)(row0 + r) * 128 + lane * 4 + j] = cM * mvv[j] + cE * evv[j];
  }
}

// ---------------- Kernel 3: fused tan -> W1(relu) -> W2 -> denorm ----------
__global__ __launch_bounds__(256) void k_head(
    const float* __restrict__ z0ws, const float* __restrict__ v0tws,
    const float* __restrict__ step_sizes, const float* __restrict__ W1,
    const float* __restrict__ b1, const float* __restrict__ W2,
    const float* __restrict__ b2, const float* __restrict__ revin_w,
    const float* __restrict__ revin_b, const float* __restrict__ mu,
    const float* __restrict__ stdv, float* __restrict__ out) {
  __shared__ float tanA[16 * 129];  // tan tile (padded)
  __shared__ float sB[32 * 256];    // W1 K-chunk; reused as pbuf[8][2][256]
  __shared__ float sH[16 * 257];    // h tile (padded)
  __shared__ float zz[2 * 128], vvs[2 * 128];
  __shared__ float dots[2][3];
  __shared__ float kX[16], kV[16];

  int t = threadIdx.x, lane = t & 31, w = t >> 5;
  int bf0 = blockIdx.x * 2;

  {  // load z0 / v0t rows (2 bf)
    int bl = t >> 7, d = t & 127;
    zz[t] = z0ws[(size_t)(bf0 + bl) * 128 + d];
    vvs[t] = v0tws[(size_t)(bf0 + bl) * 128 + d];
  }
  __syncthreads();
  if (w < 2) {  // 3 dots per bf
    float az = 0.f, av = 0.f, s = 0.f;
#pragma unroll
    for (int j = 0; j < 4; ++j) {
      float zv = zz[w * 128 + lane * 4 + j];
      float vv = vvs[w * 128 + lane * 4 + j];
      az += zv * zv;
      av += vv * vv;
      s += zv * vv;
    }
    az = wred(az);
    av = wred(av);
    s = wred(s);
    if (lane == 0) {
      dots[w][0] = az;
      dots[w][1] = av;
      dots[w][2] = s;
    }
  }
  __syncthreads();
  if (t < 16) {  // per-row scalar chain: expmap -> projx -> logmap0
    int bfl = t >> 3;
    float tt = (float)((t & 7) + 1);
    float x2 = dots[bfl][0], vv2 = dots[bfl][1], s = dots[bfl][2];
    float stp = 1.f / (1.f + expf(-step_sizes[0]));
    float st = stp * tt;
    float n = sqrtf(fmaxf(st * st * vv2, EPSF));
    float lam = 2.f / fmaxf(1.f - x2, EPSF);
    float coefS = tanhf(lam * n * 0.5f) / n * st;  // second = coefS*v0t
    float y2 = coefS * coefS * vv2;
    float xy = coefS * s;
    float A = 1.f + 2.f * xy + y2;
    float Bq = 1.f - x2;
    float den = fmaxf(1.f + 2.f * xy + x2 * y2, EPSF);
    float cX = A / den, cV = Bq * coefS / den;
    float res2 = cX * cX * x2 + 2.f * cX * cV * s + cV * cV * vv2;
    float rn = sqrtf(fmaxf(res2, EPSF));
    float ps = (rn > MAXNF) ? (MAXNF / rn) : 1.f;
    cX *= ps;
    cV *= ps;
    float n3 = rn * ps;
    float at = atanhf(fminf(n3, ATCF)) / n3;
    kX[t] = cX * at;
    kV[t] = cV * at;
  }
  __syncthreads();
#pragma unroll
  for (int q = 0; q < 8; ++q) {  // tan tile = kX*z0 + kV*v0t
    int e = q * 256 + t;
    int r = e >> 7, d = e & 127;
    int bl = r >> 3;
    tanA[r * 129 + d] = kX[r] * zz[bl * 128 + d] + kV[r] * vvs[bl * 128 + d];
  }

  // GEMM1: h(16x256) = tan(16x128) @ W1 ; relu(+b1) into LDS
  int m16 = lane & 15, koff = (lane < 16) ? 0 : 2;
  int n0A = w * 16, n0B = 128 + w * 16;
  v8f cA = {}, cB = {};
  for (int kc = 0; kc < 4; ++kc) {
    __syncthreads();
#pragma unroll
    for (int q = 0; q < 32; ++q) {
      int e = q * 256 + t;
      sB[e] = W1[kc * 32 * 256 + e];
    }
    __syncthreads();
#pragma unroll
    for (int k2 = 0; k2 < 8; ++k2) {
      int kl = k2 * 4 + koff;
      int kg = kc * 32 + kl;
      v2f a, bb;
      a.x = tanA[m16 * 129 + kg];
      a.y = tanA[m16 * 129 + kg + 1];
      bb.x = sB[kl * 256 + n0A + m16];
      bb.y = sB[(kl + 1) * 256 + n0A + m16];
      cA = wmma4(a, bb, cA);
      bb.x = sB[kl * 256 + n0B + m16];
      bb.y = sB[(kl + 1) * 256 + n0B + m16];
      cB = wmma4(a, bb, cB);
    }
  }
  __syncthreads();
#pragma unroll
  for (int r = 0; r < 8; ++r) {
    int mm = r + ((lane < 16) ? 0 : 8);
    int nA = n0A + m16, nB = n0B + m16;
    sH[mm * 257 + nA] = fmaxf(cA[r] + b1[nA], 0.f);
    sH[mm * 257 + nB] = fmaxf(cB[r] + b1[nB], 0.f);
  }
  __syncthreads();

  // GEMM2: out(16x24) = h @ W2 ; K=256 split across 8 waves, reduce in LDS
  v8f c0 = {}, c1 = {};
  int kb = w * 32;
#pragma unroll
  for (int k2 = 0; k2 < 8; ++k2) {
    int kg = kb + k2 * 4 + koff;
    v2f a, b0f, b1f;
    a.x = sH[m16 * 257 + kg];
    a.y = sH[m16 * 257 + kg + 1];
    int nn0 = m16, nn1 = 16 + m16;
    b0f.x = W2[kg * 24 + nn0];
    b0f.y = W2[(kg + 1) * 24 + nn0];
    b1f.x = (nn1 < 24) ? W2[kg * 24 + nn1] : 0.f;
    b1f.y = (nn1 < 24) ? W2[(kg + 1) * 24 + nn1] : 0.f;
    c0 = wmma4(a, b0f, c0);
    c1 = wmma4(a, b1f, c1);
  }
  float* pbuf = sB;  // 8 waves x 2 tiles x 256 floats
#pragma unroll
  for (int r = 0; r < 8; ++r) {
    pbuf[w * 512 + r * 32 + lane] = c0[r];
    pbuf[w * 512 + 256 + r * 32 + lane] = c1[r];
  }
  __syncthreads();

  for (int e = t; e < 16 * 24; e += 256) {
    int mrow = e / 24, n = e % 24;
    int tl = n >> 4, col = n & 15;
    int off = tl * 256 + (mrow & 7) * 32 + col + ((mrow >= 8) ? 16 : 0);
    float acc = b2[n];
#pragma unroll
    for (int ww = 0; ww < 8; ++ww) acc += pbuf[ww * 512 + off];
    int bf = bf0 + (mrow >> 3);
    int ti = mrow & 7;
    int b = bf >> 7, f = bf & 127;
    int l2 = ti * 24 + n;
    float val = (acc - revin_b[f]) / revin_w[f] * stdv[bf] + mu[bf];
    out[(size_t)b * (192 * 128) + (size_t)l2 * 128 + f] = val;
  }
}

// ---------------------------------------------------------------------------
extern "C" void kernel_launch(void* const* d_in, const int* in_sizes, int n_in,
                              void* d_out, int out_size, void* d_ws,
                              size_t ws_size, hipStream_t stream) {
  const float* x = (const float*)d_in[0];
  const float* revin_w = (const float*)d_in[1];
  const float* revin_b = (const float*)d_in[2];
  const float* W_enc = (const float*)d_in[3];
  const float* b_enc = (const float*)d_in[4];
  const float* W_vel = (const float*)d_in[5];
  const float* b_vel = (const float*)d_in[6];
  const float* step_sizes = (const float*)d_in[7];
  const float* W1 = (const float*)d_in[8];
  const float* b1 = (const float*)d_in[9];
  const float* W2 = (const float*)d_in[10];
  const float* b2 = (const float*)d_in[11];
  float* out = (float*)d_out;

  float* ws = (float*)d_ws;
  float* mu = ws;                        // 16384
  float* stdv = ws + 16384;              // 16384
  float* vinit = ws + 32768;             // 16384*128
  float* z0 = vinit + 16384 * 128;       // 16384*128
  float* v0t = z0 + 16384 * 128;         // 16384*128

  k_stats<<<128, 128, 0, stream>>>(x, mu, stdv);
  k_encode<<<16384, 128, 0, stream>>>(x, revin_w, revin_b, W_enc, b_enc, mu,
                                      stdv, vinit, z0);
  k_v0t<<<1024, 256, 0, stream>>>(vinit, W_vel, b_vel, v0t);
  k_head<<<8192, 256, 0, stream>>>(z0, v0t, step_sizes, W1, b1, W2, b2,
                                   revin_w, revin_b, mu, stdv, out);
}